// A4Quantizer_35021163331725
// MI455X (gfx1250) — compile-verified
//
#include <hip/hip_runtime.h>
#include <stdint.h>

#define ROWLEN   8192          // columns per row (= quant dim)
#define QBLOCK   32            // MXFP4 block size == wave32
#define THREADS  256           // 8 waves per workgroup
#define WAVES    (THREADS / 32)

// DPP-based partial max: combine v with a DPP-permuted copy (pure VALU).
#define DPP_FMAX(v, ctrl)                                                     \
    fmaxf((v), __int_as_float(__builtin_amdgcn_update_dpp(                    \
                   __float_as_int(v), __float_as_int(v), (ctrl),              \
                   0xF, 0xF, true)))

// Full wave32 max reduction: 4 DPP steps + 1 permlanex16 (no DS traffic).
__device__ __forceinline__ float wave32_max(float a) {
    a = DPP_FMAX(a, 0xB1);   // quad_perm [1,0,3,2] : pairs lane^1
    a = DPP_FMAX(a, 0x4E);   // quad_perm [2,3,0,1] : pairs lane^2
    a = DPP_FMAX(a, 0x141);  // row_half_mirror     : pairs across 4s within 8
    a = DPP_FMAX(a, 0x140);  // row_mirror          : pairs across 8s within 16
    int ai = __float_as_int(a);
    // identity nibble selects -> lane i reads lane i^16 (cross 16-lane rows)
    int t = __builtin_amdgcn_permlanex16(ai, ai, 0x76543210, 0xfedcba98,
                                         false, false);
    return fmaxf(a, __int_as_float(t));
}

// One workgroup per matrix row:
//  1) async-copy the 32KB row into LDS (global_load_async_to_lds_b128, NT)
//  2) each wave quantizes one 32-elem block per step: gather from LDS via
//     perm, VALU wave-max, fp4 fake-quant, coalesced 128B NT store.
__global__ __launch_bounds__(THREADS) void mxfp4_perm_quant_kernel(
    const float* __restrict__ x,
    const int*   __restrict__ perm,
    float*       __restrict__ out)
{
    __shared__ float row[ROWLEN];

    const int      tid = threadIdx.x;
    const size_t   r   = blockIdx.x;
    const float*   src = x   + r * (size_t)ROWLEN;
    float*         dst = out + r * (size_t)ROWLEN;

    // ---- Stage row -> LDS with async copies (16B per lane per issue) ----
    #pragma unroll
    for (int i = 0; i < ROWLEN / (THREADS * 4); ++i) {
        const int idx = (i * THREADS + tid) * 4;
        // low 32 bits of a generic LDS pointer == LDS byte offset
        uint32_t loff = (uint32_t)(uintptr_t)&row[idx];
        uint64_t ga   = (uint64_t)(uintptr_t)(src + idx);
        asm volatile("global_load_async_to_lds_b128 %0, %1, off th:TH_LOAD_NT"
                     :: "v"(loff), "v"(ga) : "memory");
    }
    asm volatile("s_wait_asynccnt 0" ::: "memory");
    __syncthreads();

    // ---- Quantize: one 32-element block per wave per iteration ----
    const int lane = tid & 31;
    const int wid  = tid >> 5;

    #pragma unroll 4
    for (int b = wid; b < ROWLEN / QBLOCK; b += WAVES) {
        const int j = b * QBLOCK + lane;

        const int   p = perm[j];       // hot in L2 (shared by all rows)
        const float v = row[p];        // random LDS read (only DS op here)

        // block amax across the wave (wave32 == one quant block)
        const float a = wave32_max(fabsf(v));

        // shared power-of-two scale: 2^(floor(log2(amax)) - 2); 1.0 if amax==0
        int k = 0;
        if (a > 0.0f) {
            int e;
            (void)frexpf(a, &e);       // a = f * 2^e, f in [0.5, 1)
            k = (e - 1) - 2;           // floor(log2(a)) - 2
        }
        const float scale = ldexpf(1.0f, k);

        // y = clip(v / scale, -6, 6)   (exact power-of-2 scaling)
        float y = ldexpf(v, -k);
        y = fminf(fmaxf(y, -6.0f), 6.0f);

        // nearest fp4 magnitude; midpoints [.25,.75,1.25,1.75,2.5,3.5,5],
        // ties round up (searchsorted side="right")
        const float m = fabsf(y);
        float q;
        if      (m < 0.25f) q = 0.0f;
        else if (m < 0.75f) q = 0.5f;
        else if (m < 1.25f) q = 1.0f;
        else if (m < 1.75f) q = 1.5f;
        else if (m < 2.5f)  q = 2.0f;
        else if (m < 3.5f)  q = 3.0f;
        else if (m < 5.0f)  q = 4.0f;
        else                q = 6.0f;

        const float o = copysignf(q, y) * scale;
        __builtin_nontemporal_store(o, dst + j);   // 128B/wave write-once stream
    }
}

extern "C" void kernel_launch(void* const* d_in, const int* in_sizes, int n_in,
                              void* d_out, int out_size, void* d_ws, size_t ws_size,
                              hipStream_t stream) {
    const float* x    = (const float*)d_in[0];
    const int*   perm = (const int*)d_in[1];
    float*       out  = (float*)d_out;

    const int rows = in_sizes[0] / ROWLEN;   // 8192
    mxfp4_perm_quant_kernel<<<dim3(rows), dim3(THREADS), 0, stream>>>(x, perm, out);
}